// GNNFeatureExtraction_87591563035159
// MI455X (gfx1250) — compile-verified
//
#include <hip/hip_runtime.h>
#include <hip/hip_bf16.h>

typedef __attribute__((ext_vector_type(2))) float v2f;
typedef __attribute__((ext_vector_type(8))) float v8f;

#define FDIM 128

// ---------------- degree / normalization ----------------
__global__ __launch_bounds__(256) void k_deg_init(float* __restrict__ deg, int N) {
    int i = blockIdx.x * 256 + threadIdx.x;
    if (i < N) deg[i] = 1.0f;   // +1 self-loop
}

__global__ __launch_bounds__(256) void k_deg_count(const int* __restrict__ dst,
                                                   float* __restrict__ deg, int E) {
    int i = blockIdx.x * 256 + threadIdx.x;
    if (i < E) unsafeAtomicAdd(&deg[dst[i]], 1.0f);   // global_atomic_add_f32
}

__global__ __launch_bounds__(256) void k_deg_rsqrt(float* __restrict__ deg, int N) {
    int i = blockIdx.x * 256 + threadIdx.x;
    if (i < N) deg[i] = rsqrtf(deg[i]);               // deg buffer becomes dinv
}

// ---------------- WMMA GEMM: H = X @ W  (fp32, 16x16x4) ----------------
// Block = 256 threads = 8 waves. Block computes a 16-row x 128-col tile of H.
// Wave w computes the 16x16 tile at columns [16w, 16w+16).
__global__ __launch_bounds__(256) void k_gemm_wmma(const float* __restrict__ X,
                                                   const float* __restrict__ W,
                                                   float* __restrict__ H, int N) {
    __shared__ float xs[16 * FDIM];                   // 8 KB x-tile
    const int row0 = blockIdx.x * 16;
    const int tid  = threadIdx.x;

    // coalesced load of the 16x128 x-tile into LDS (zero-pad tail rows)
    for (int i = tid; i < 16 * FDIM; i += 256) {
        int r = i >> 7, c = i & (FDIM - 1);
        xs[i] = (row0 + r < N) ? X[(size_t)(row0 + r) * FDIM + c] : 0.0f;
    }
    __syncthreads();

    const int wave   = tid >> 5;        // 0..7 -> column tile
    const int lane   = tid & 31;
    const int laneLo = lane & 15;       // M (for A) / N (for B)
    const int hi     = lane >> 4;       // K-half selector
    const int col0   = wave * 16;

    v8f acc = {};
    for (int k = 0; k < FDIM; k += 4) {
        const int ka = k + hi * 2;      // lanes 0-15: K=k,k+1 ; lanes 16-31: K=k+2,k+3
        v2f a, b;
        // A fragment: lane holds row M=laneLo of X-tile
        a.x = xs[laneLo * FDIM + ka];
        a.y = xs[laneLo * FDIM + ka + 1];
        // B fragment: lane holds column N=col0+laneLo of W
        b.x = W[(size_t)ka       * FDIM + col0 + laneLo];
        b.y = W[(size_t)(ka + 1) * FDIM + col0 + laneLo];
        acc = __builtin_amdgcn_wmma_f32_16x16x4_f32(
            /*neg_a=*/false, a, /*neg_b=*/false, b,
            /*c_mod=*/(short)0, acc, /*reuse_a=*/false, /*reuse_b=*/false);
    }

    // D layout: lane holds column N=col0+laneLo, rows M = hi*8 + r
#pragma unroll
    for (int r = 0; r < 8; ++r) {
        int row = row0 + hi * 8 + r;
        if (row < N) H[(size_t)row * FDIM + col0 + laneLo] = acc[r];
    }
}

// ---------------- self-loop init: OUT = H * dinv^2 ----------------
__global__ __launch_bounds__(256) void k_selfloop(const float* __restrict__ H,
                                                  const float* __restrict__ dinv,
                                                  float* __restrict__ OUT, int N) {
    int i = blockIdx.x * 256 + threadIdx.x;         // over N*32 float4s
    if (i < N * (FDIM / 4)) {
        int row = i >> 5;
        float di = dinv[row];
        float s  = di * di;
        float4 v = ((const float4*)H)[i];
        v.x *= s; v.y *= s; v.z *= s; v.w *= s;
        ((float4*)OUT)[i] = v;
    }
}

// ---------------- edge scatter: OUT[dst] += H[src] * dinv[src]*dinv[dst] ----------------
// one wave per edge; each lane moves a float4 (32 lanes * 16B = 512B = full row)
__global__ __launch_bounds__(256) void k_scatter(const float* __restrict__ H,
                                                 const int* __restrict__ src,
                                                 const int* __restrict__ dst,
                                                 const float* __restrict__ dinv,
                                                 float* __restrict__ OUT, int E) {
    int e = blockIdx.x * 8 + (threadIdx.x >> 5);
    if (e >= E) return;
    int lane = threadIdx.x & 31;
    int s = src[e], d = dst[e];
    float nrm = dinv[s] * dinv[d];
    float4 v = ((const float4*)(H + (size_t)s * FDIM))[lane];
    float* o = OUT + (size_t)d * FDIM + lane * 4;
    unsafeAtomicAdd(o + 0, v.x * nrm);
    unsafeAtomicAdd(o + 1, v.y * nrm);
    unsafeAtomicAdd(o + 2, v.z * nrm);
    unsafeAtomicAdd(o + 3, v.w * nrm);
}

// ---------------- epilogue: OUT = (relu?) (OUT + b) ----------------
__global__ __launch_bounds__(256) void k_bias_act(float* __restrict__ OUT,
                                                  const float* __restrict__ b,
                                                  int total, int do_relu) {
    int i = blockIdx.x * 256 + threadIdx.x;
    if (i < total) {
        float v = OUT[i] + b[i & (FDIM - 1)];
        OUT[i] = do_relu ? fmaxf(v, 0.0f) : v;
    }
}

// ---------------- one GCN layer ----------------
static void gcn_layer(const float* in, const float* W, const float* b,
                      const int* src, const int* dst, const float* dinv,
                      float* G /*gemm buf*/, float* out,
                      int N, int E, int do_relu, hipStream_t stream) {
    int total = N * FDIM;
    k_gemm_wmma <<<(N + 15) / 16, 256, 0, stream>>>(in, W, G, N);
    k_selfloop  <<<(N * (FDIM / 4) + 255) / 256, 256, 0, stream>>>(G, dinv, out, N);
    k_scatter   <<<(E + 7) / 8, 256, 0, stream>>>(G, src, dst, dinv, out, E);
    k_bias_act  <<<(total + 255) / 256, 256, 0, stream>>>(out, b, total, do_relu);
}

extern "C" void kernel_launch(void* const* d_in, const int* in_sizes, int n_in,
                              void* d_out, int out_size, void* d_ws, size_t ws_size,
                              hipStream_t stream) {
    const float* x   = (const float*)d_in[0];
    const int*   ei  = (const int*)d_in[1];     // [2, E] int32 (JAX default x64-off)
    const float* W1  = (const float*)d_in[2];
    const float* b1  = (const float*)d_in[3];
    const float* W2  = (const float*)d_in[4];
    const float* b2  = (const float*)d_in[5];
    const float* W3  = (const float*)d_in[6];
    const float* b3  = (const float*)d_in[7];

    const int N = in_sizes[0] / FDIM;
    const int E = in_sizes[1] / 2;
    const int* src = ei;
    const int* dst = ei + E;

    // workspace layout: dinv | G (N*128) | H (N*128)
    float* dinv = (float*)d_ws;
    size_t nPad = ((size_t)N + 255) & ~(size_t)255;
    float* G = dinv + nPad;
    float* H = G + (size_t)N * FDIM;

    // normalization: dinv = rsqrt(deg + 1)
    k_deg_init  <<<(N + 255) / 256, 256, 0, stream>>>(dinv, N);
    k_deg_count <<<(E + 255) / 256, 256, 0, stream>>>(dst, dinv, E);
    k_deg_rsqrt <<<(N + 255) / 256, 256, 0, stream>>>(dinv, N);

    // layer 1: x -> H (relu)
    gcn_layer(x, W1, b1, src, dst, dinv, G, H, N, E, /*relu=*/1, stream);
    // layer 2: H -> H (gemm consumes H into G first, so overwrite is safe)
    gcn_layer(H, W2, b2, src, dst, dinv, G, H, N, E, /*relu=*/1, stream);
    // layer 3: H -> d_out (no relu)
    gcn_layer(H, W3, b3, src, dst, dinv, G, (float*)d_out, N, E, /*relu=*/0, stream);
}